// MiniMaxMoE_59803124630218
// MI455X (gfx1250) — compile-verified
//
#include <hip/hip_runtime.h>
#include <cstdint>

// MiniMax-style top-2 MoE (B=2,S=1024,H=2048,F=4096,E=8) for gfx1250.
// fp32 throughout; matrix cores via V_WMMA_F32_16X16X4_F32; x-tile staging via
// GLOBAL_LOAD_ASYNC_TO_LDS_B128 (ASYNCcnt); weight streams prefetched.

#define H_DIM  2048
#define F_DIM  4096
#define E_NUM  8
#define TOPK   2
#define TILE_M 16          // tokens per workgroup tile
#define F_BLK  128         // F columns per block (8 waves x 16)
#define K_BLK  512         // H staged in LDS per chunk
#define NT     16          // 16x16 n-tiles per wave in down-proj (8 waves * 256 = 2048 = H_DIM)

typedef float v2f __attribute__((ext_vector_type(2)));
typedef float v8f __attribute__((ext_vector_type(8)));

// exact pointee type the async-LDS builtin expects: int vector_size(16)
typedef int v4i_vs __attribute__((vector_size(16)));
typedef v4i_vs __attribute__((address_space(1)))* g_b128p;   // global
typedef v4i_vs __attribute__((address_space(3)))* l_b128p;   // LDS

// ---------------------------------------------------------------- init
__global__ void moe_init(float* __restrict__ out, int out_n,
                         int* __restrict__ rows, float* __restrict__ rws,
                         int* __restrict__ scal, int nscal, int max_slots) {
    int i = blockIdx.x * blockDim.x + threadIdx.x;
    if (i < out_n) out[i] = 0.0f;
    if (i < max_slots) { rows[i] = -1; rws[i] = 0.0f; }
    if (i < nscal) scal[i] = 0;
}

// ---------------------------------------------------------------- router
// One wave32 per token: logits over E=8 experts, softmax, top-2, renorm.
__global__ void moe_router(const float* __restrict__ x, const float* __restrict__ gw,
                           int* __restrict__ topk_i, float* __restrict__ topk_w,
                           int* __restrict__ counts, int T) {
    int wave = threadIdx.x >> 5;
    int lane = threadIdx.x & 31;
    int t = blockIdx.x * (blockDim.x >> 5) + wave;
    if (t >= T) return;
    const float* xt = x + (size_t)t * H_DIM;
    float acc[E_NUM];
#pragma unroll
    for (int e = 0; e < E_NUM; ++e) acc[e] = 0.0f;
    for (int k = lane; k < H_DIM; k += 32) {
        float xv = xt[k];
#pragma unroll
        for (int e = 0; e < E_NUM; ++e) acc[e] += xv * gw[e * H_DIM + k];
    }
#pragma unroll
    for (int e = 0; e < E_NUM; ++e) {
#pragma unroll
        for (int off = 16; off > 0; off >>= 1)
            acc[e] += __shfl_xor(acc[e], off, 32);
    }
    if (lane == 0) {
        float mx = acc[0];
#pragma unroll
        for (int e = 1; e < E_NUM; ++e) mx = fmaxf(mx, acc[e]);
        float p[E_NUM], s = 0.0f;
#pragma unroll
        for (int e = 0; e < E_NUM; ++e) { p[e] = __expf(acc[e] - mx); s += p[e]; }
        float inv = 1.0f / s;
#pragma unroll
        for (int e = 0; e < E_NUM; ++e) p[e] *= inv;
        int i0 = 0;
#pragma unroll
        for (int e = 1; e < E_NUM; ++e) if (p[e] > p[i0]) i0 = e;
        int i1 = (i0 == 0) ? 1 : 0;
#pragma unroll
        for (int e = 0; e < E_NUM; ++e) if (e != i0 && p[e] > p[i1]) i1 = e;
        float sw = p[i0] + p[i1];
        topk_i[t * 2 + 0] = i0;  topk_w[t * 2 + 0] = p[i0] / sw;
        topk_i[t * 2 + 1] = i1;  topk_w[t * 2 + 1] = p[i1] / sw;
        atomicAdd(&counts[i0], 1);
        atomicAdd(&counts[i1], 1);
    }
}

// ---------------------------------------------------------------- scan (E=8, trivial)
__global__ void moe_scan(const int* __restrict__ counts, int* __restrict__ offs) {
    int o = 0;
    for (int e = 0; e < E_NUM; ++e) {
        offs[e] = o;
        int c = counts[e];
        o += ((c + TILE_M - 1) / TILE_M) * TILE_M;   // pad each segment to tile
    }
    offs[E_NUM] = o;
}

// ---------------------------------------------------------------- scatter
__global__ void moe_scatter(const int* __restrict__ topk_i, const float* __restrict__ topk_w,
                            const int* __restrict__ offs, int* __restrict__ fill,
                            int* __restrict__ rows, float* __restrict__ rws, int T) {
    int t = blockIdx.x * blockDim.x + threadIdx.x;
    if (t >= T) return;
#pragma unroll
    for (int k = 0; k < TOPK; ++k) {
        int e = topk_i[t * 2 + k];
        int pos = atomicAdd(&fill[e], 1);
        int slot = offs[e] + pos;
        rows[slot] = t;
        rws[slot]  = topk_w[t * 2 + k];
    }
}

// ---------------------------------------------------------------- grouped expert FFN
// 8 waves / 256 threads per 16-token tile. Wave w owns h-tile cols [fb+16w,fb+16w+16)
// and y-accumulator cols [256w, 256w+256) (16 x v8f in registers).
__global__ __launch_bounds__(256)
void moe_ffn(const float* __restrict__ x,
             const float* __restrict__ w_gate,
             const float* __restrict__ w_up,
             const float* __restrict__ w_down,
             const int*   __restrict__ offs,
             const int*   __restrict__ rows,
             const float* __restrict__ rws,
             float*       __restrict__ out) {
    __shared__ float xs[TILE_M][K_BLK + 4];    // ~33 KB  x chunk  [16 x 512]
    __shared__ float hb[TILE_M][F_BLK + 4];    // ~8.4 KB h block  [16 x 128]
    __shared__ int   s_row[TILE_M];
    __shared__ float s_w[TILE_M];

    const int slot0 = blockIdx.x * TILE_M;
    if (slot0 >= offs[E_NUM]) return;          // inactive padded tile (uniform)
    int e = 0;
    for (int i = 0; i < E_NUM; ++i)
        if (slot0 >= offs[i] && slot0 < offs[i + 1]) e = i;

    if (threadIdx.x < TILE_M) {
        s_row[threadIdx.x] = rows[slot0 + threadIdx.x];
        s_w[threadIdx.x]   = rws[slot0 + threadIdx.x];
    }
    __syncthreads();

    const int lane = threadIdx.x & 31;
    const int wave = threadIdx.x >> 5;
    const int half = lane >> 4;      // 0: lanes 0-15 (K0,K1) / 1: lanes 16-31 (K2,K3)
    const int l16  = lane & 15;

    const float* Wg = w_gate + (size_t)e * F_DIM * H_DIM;
    const float* Wu = w_up   + (size_t)e * F_DIM * H_DIM;
    const float* Wd = w_down + (size_t)e * H_DIM * F_DIM;

    v8f zero = {};
    v8f yacc[NT];
#pragma unroll
    for (int i = 0; i < NT; ++i) yacc[i] = zero;

    for (int fb = 0; fb < F_DIM; fb += F_BLK) {
        // ---- gate/up: this wave computes g,u for cols [fcol, fcol+16) over K=H
        const int fcol = fb + wave * 16;
        v8f gacc = zero, uacc = zero;
        for (int kc = 0; kc < H_DIM; kc += K_BLK) {
            __syncthreads();
            // async stage of x[rows, kc:kc+K_BLK] into LDS (pad rows clamp to row 0;
            // their results are computed but never stored). b128 per lane, no VGPR
            // round-trip, tracked by ASYNCcnt.
            for (int idx = threadIdx.x; idx < TILE_M * (K_BLK / 4); idx += 256) {
                int m   = idx / (K_BLK / 4);
                int kk4 = (idx % (K_BLK / 4)) * 4;
                int r   = s_row[m];
                if (r < 0) r = 0;
                const float* gsrc = x + (size_t)r * H_DIM + kc + kk4;
                float*       ldst = &xs[m][kk4];
                __builtin_amdgcn_global_load_async_to_lds_b128(
                    (g_b128p)(uintptr_t)gsrc,
                    (l_b128p)(uint32_t)(uintptr_t)ldst, 0, 0);
            }
            __builtin_amdgcn_s_wait_asynccnt(0);
            __syncthreads();

            const float* WgP = Wg + (size_t)(fcol + l16) * H_DIM + kc + 2 * half;
            const float* WuP = Wu + (size_t)(fcol + l16) * H_DIM + kc + 2 * half;
            if (kc + K_BLK < H_DIM) {            // pull next K-chunk of B toward L2/L0
                __builtin_prefetch(WgP + K_BLK, 0, 1);
                __builtin_prefetch(WuP + K_BLK, 0, 1);
            }
#pragma unroll 4
            for (int kk = 0; kk < K_BLK; kk += 4) {
                v2f a  = *(const v2f*)&xs[l16][kk + 2 * half];
                v2f bg = *(const v2f*)(WgP + kk);
                v2f bu = *(const v2f*)(WuP + kk);
                gacc = __builtin_amdgcn_wmma_f32_16x16x4_f32(false, a, false, bg,
                                                             (short)0, gacc, false, false);
                uacc = __builtin_amdgcn_wmma_f32_16x16x4_f32(false, a, false, bu,
                                                             (short)0, uacc, false, false);
            }
        }
        // ---- h = silu(g) * u  -> LDS
        __syncthreads();
#pragma unroll
        for (int r = 0; r < 8; ++r) {
            int m = r + 8 * half;
            float g = gacc[r], u = uacc[r];
            hb[m][wave * 16 + l16] = (g / (1.0f + __expf(-g))) * u;
        }
        __syncthreads();
        // ---- down-proj: yacc[nt] += h[16 x 128] * Wd[cols, fb:fb+128]^T
#pragma unroll
        for (int nt = 0; nt < NT; ++nt) {
            v8f acc = yacc[nt];
            const float* WdP = Wd + (size_t)(wave * 256 + nt * 16 + l16) * F_DIM + fb + 2 * half;
#pragma unroll 4
            for (int k = 0; k < F_BLK; k += 4) {
                v2f a = *(const v2f*)&hb[l16][k + 2 * half];
                v2f b = *(const v2f*)(WdP + k);
                acc = __builtin_amdgcn_wmma_f32_16x16x4_f32(false, a, false, b,
                                                            (short)0, acc, false, false);
            }
            yacc[nt] = acc;
        }
    }

    // ---- scale by routing weight and scatter-add (2 commutative adds per element)
#pragma unroll
    for (int nt = 0; nt < NT; ++nt) {
        int col = wave * 256 + nt * 16 + l16;
#pragma unroll
        for (int r = 0; r < 8; ++r) {
            int m = r + 8 * half;
            int tok = s_row[m];
            if (tok >= 0)
                atomicAdd(&out[(size_t)tok * H_DIM + col], yacc[nt][r] * s_w[m]);
        }
    }
}

// ---------------------------------------------------------------- launch
extern "C" void kernel_launch(void* const* d_in, const int* in_sizes, int n_in,
                              void* d_out, int out_size, void* d_ws, size_t ws_size,
                              hipStream_t stream) {
    const float* x  = (const float*)d_in[0];   // [B,S,H]
    const float* gw = (const float*)d_in[1];   // [E,H]
    const float* wg = (const float*)d_in[2];   // [E,F,H]
    const float* wu = (const float*)d_in[3];   // [E,F,H]
    const float* wd = (const float*)d_in[4];   // [E,H,F]
    float* out = (float*)d_out;

    const int T = in_sizes[0] / H_DIM;                         // 2048 tokens
    const int max_slots = T * TOPK + E_NUM * (TILE_M - 1);     // worst-case padded rows
    const int max_tiles = (max_slots + TILE_M - 1) / TILE_M;

    // workspace layout
    int*   counts = (int*)d_ws;                 // [E]
    int*   fill   = counts + E_NUM;             // [E]
    int*   offs   = fill + E_NUM;               // [E+1]
    int*   topk_i = offs + (E_NUM + 1);         // [T*2]
    float* topk_w = (float*)(topk_i + 2 * T);   // [T*2]
    int*   rows   = (int*)(topk_w + 2 * T);     // [max_slots]
    float* rwsP   = (float*)(rows + max_slots); // [max_slots]
    const int nscal = 2 * E_NUM + (E_NUM + 1);

    int initN = out_size > max_slots ? out_size : max_slots;
    moe_init<<<(initN + 255) / 256, 256, 0, stream>>>(out, out_size, rows, rwsP,
                                                      counts, nscal, max_slots);
    moe_router<<<(T + 7) / 8, 256, 0, stream>>>(x, gw, topk_i, topk_w, counts, T);
    moe_scan<<<1, 1, 0, stream>>>(counts, offs);
    moe_scatter<<<(T + 255) / 256, 256, 0, stream>>>(topk_i, topk_w, offs, fill,
                                                     rows, rwsP, T);
    moe_ffn<<<max_tiles, 256, 0, stream>>>(x, wg, wu, wd, offs, rows, rwsP, out);
}